// Net_70188355551324
// MI455X (gfx1250) — compile-verified
//
#include <hip/hip_runtime.h>

typedef __attribute__((ext_vector_type(2))) float v2f;
typedef __attribute__((ext_vector_type(8))) float v8f;

#define HID 64
#define BN_EPS 1e-5f

// ---------------------------------------------------------------------------
// Edge scatter-add: agg[dst] += h[src]  (agg pre-initialized with h itself,
// giving h_i + sum_{j->i} h_j).  One thread handles a float4 chunk of one
// edge's feature row; consecutive threads touch consecutive features of the
// same edge -> coalesced gather, 4 global_atomic_add_f32 per thread.
// ---------------------------------------------------------------------------
__global__ void edge_scatter(const float* __restrict__ h, float* __restrict__ agg,
                             const int* __restrict__ ei, int E, int FIN, int lc /* log2(FIN/4) */) {
    long long idx = (long long)blockIdx.x * blockDim.x + threadIdx.x;
    long long total = (long long)E << lc;
    if (idx >= total) return;
    int e = (int)(idx >> lc);
    int q = (int)(idx & ((1 << lc) - 1));
    int s = ei[e];       // src node
    int d = ei[E + e];   // dst node
    const float4* xs = reinterpret_cast<const float4*>(h + (long long)s * FIN) + q;
    float4 v = *xs;
    float* ap = agg + (long long)d * FIN + q * 4;
    atomicAdd(ap + 0, v.x);
    atomicAdd(ap + 1, v.y);
    atomicAdd(ap + 2, v.z);
    atomicAdd(ap + 3, v.w);
}

// ---------------------------------------------------------------------------
// Fused 2-layer MLP on a 16-node tile using V_WMMA_F32_16X16X4_F32.
// One wave (32 threads) per block; grid = ceil(N/16) tiles, no early exit -> EXEC all-ones.
//   T  = relu(A(16xFIN) @ W1(FINx64) + b1)   (kept in LDS, row-major 16x64)
//   H  = relu(T @ W2(64x64) + b2)            (written to global h)
// Fragment layouts per CDNA5 ISA 7.12.2:
//   A 16x4 : lane l -> row l%16, K-pair 2*(l/16)           (2 VGPRs)
//   B 4x16 : lane l -> col l%16, K-pair 2*(l/16)           (2 VGPRs)
//   C 16x16: VGPR r -> row r + 8*(l/16), col l%16          (8 VGPRs)
// ---------------------------------------------------------------------------
template <int FIN>
__global__ void mlp_wmma(const float* __restrict__ agg,
                         const float* __restrict__ w1, const float* __restrict__ b1,
                         const float* __restrict__ w2, const float* __restrict__ b2,
                         float* __restrict__ h, int N) {
    __shared__ float T[16 * HID];
    const int lane = threadIdx.x;        // blockDim.x == 32 (wave32)
    const int l16  = lane & 15;
    const int lhi  = lane >> 4;          // 0 or 1
    const int row0 = blockIdx.x * 16;

    // ---------------- GEMM1: agg-tile @ W1 ----------------
    v8f acc[4] = {};
    const int arow = min(row0 + l16, N - 1);
    const float* __restrict__ aptr = agg + (long long)arow * FIN + 2 * lhi;
    for (int k = 0; k < FIN; k += 4) {
        v2f a;
        a.x = aptr[k];
        a.y = aptr[k + 1];
        const float* __restrict__ wk = w1 + (k + 2 * lhi) * HID + l16;
#pragma unroll
        for (int nt = 0; nt < 4; ++nt) {
            v2f b;
            b.x = wk[nt * 16];
            b.y = wk[HID + nt * 16];
            acc[nt] = __builtin_amdgcn_wmma_f32_16x16x4_f32(
                false, a, false, b, (short)0, acc[nt], false, false);
        }
    }
    // bias + relu, stash T in LDS (row-major) for re-fragmenting as A of GEMM2
#pragma unroll
    for (int nt = 0; nt < 4; ++nt) {
        int col = nt * 16 + l16;
        float bias = b1[col];
#pragma unroll
        for (int r = 0; r < 8; ++r) {
            int m = r + 8 * lhi;
            float v = acc[nt][r] + bias;
            T[m * HID + col] = v > 0.0f ? v : 0.0f;
        }
    }
    __syncthreads();

    // ---------------- GEMM2: T @ W2 ----------------
    v8f acc2[4] = {};
    const float* __restrict__ tptr = T + l16 * HID + 2 * lhi;
    for (int k = 0; k < HID; k += 4) {
        v2f a;
        a.x = tptr[k];
        a.y = tptr[k + 1];
        const float* __restrict__ wk = w2 + (k + 2 * lhi) * HID + l16;
#pragma unroll
        for (int nt = 0; nt < 4; ++nt) {
            v2f b;
            b.x = wk[nt * 16];
            b.y = wk[HID + nt * 16];
            acc2[nt] = __builtin_amdgcn_wmma_f32_16x16x4_f32(
                false, a, false, b, (short)0, acc2[nt], false, false);
        }
    }
#pragma unroll
    for (int nt = 0; nt < 4; ++nt) {
        int col = nt * 16 + l16;
        float bias = b2[col];
#pragma unroll
        for (int r = 0; r < 8; ++r) {
            int m = r + 8 * lhi;
            if (row0 + m < N) {
                float v = acc2[nt][r] + bias;
                h[(long long)(row0 + m) * HID + col] = v > 0.0f ? v : 0.0f;
            }
        }
    }
}

// ---------------------------------------------------------------------------
// BatchNorm batch statistics: stats[0..63] = sum, stats[64..127] = sum of sq.
// 256 threads/block -> 4 row-streams of 64 channels, LDS reduce, 128 atomics.
// ---------------------------------------------------------------------------
__global__ void bn_stats(const float* __restrict__ h, float* __restrict__ stats, int N) {
    __shared__ float ls[256];
    __shared__ float lss[256];
    int tid = threadIdx.x;
    int c = tid & 63;
    int g = tid >> 6;                   // 0..3
    float s = 0.0f, ss = 0.0f;
    for (int r = blockIdx.x * 4 + g; r < N; r += gridDim.x * 4) {
        float v = h[(long long)r * HID + c];
        s += v;
        ss += v * v;
    }
    ls[tid] = s;
    lss[tid] = ss;
    __syncthreads();
    if (tid < 64) {
        float a = ls[tid] + ls[tid + 64] + ls[tid + 128] + ls[tid + 192];
        float b = lss[tid] + lss[tid + 64] + lss[tid + 128] + lss[tid + 192];
        atomicAdd(&stats[tid], a);
        atomicAdd(&stats[64 + tid], b);
    }
}

// h = relu((h - mu) * rsqrt(var + eps) * gamma + beta), biased variance.
__global__ void bn_apply(float* __restrict__ h, const float* __restrict__ stats,
                         const float* __restrict__ gamma, const float* __restrict__ beta, int N) {
    long long idx = (long long)blockIdx.x * blockDim.x + threadIdx.x;
    if (idx >= (long long)N * HID) return;
    int c = (int)(idx & 63);
    float invN = 1.0f / (float)N;
    float mu  = stats[c] * invN;
    float var = stats[64 + c] * invN - mu * mu;
    float v = (h[idx] - mu) * rsqrtf(var + BN_EPS) * gamma[c] + beta[c];
    h[idx] = v > 0.0f ? v : 0.0f;
}

// pooled[batch[n]] += h[n]
__global__ void pool_scatter(const float* __restrict__ h, const int* __restrict__ batch,
                             float* __restrict__ pooled, int N) {
    long long idx = (long long)blockIdx.x * blockDim.x + threadIdx.x;
    if (idx >= (long long)N * HID) return;
    int n = (int)(idx >> 6);
    int c = (int)(idx & 63);
    atomicAdd(&pooled[(long long)batch[n] * HID + c], h[idx]);
}

// out[g] = pooled[g] . fc_w + fc_b
__global__ void pool_gemv(const float* __restrict__ pooled, const float* __restrict__ fcw,
                          const float* __restrict__ fcb, float* __restrict__ out, int G) {
    int g = blockIdx.x * blockDim.x + threadIdx.x;
    if (g >= G) return;
    float s = fcb[0];
#pragma unroll 8
    for (int c = 0; c < HID; ++c) s += pooled[g * HID + c] * fcw[c];
    out[g] = s;
}

// ---------------------------------------------------------------------------
extern "C" void kernel_launch(void* const* d_in, const int* in_sizes, int n_in,
                              void* d_out, int out_size, void* d_ws, size_t ws_size,
                              hipStream_t stream) {
    (void)n_in; (void)ws_size;
    const float* x    = (const float*)d_in[0];
    const int*   ei   = (const int*)d_in[1];
    const int*   batch= (const int*)d_in[2];
    const float* w1[3], *b1[3], *w2[3], *b2[3], *gm[3], *bt[3];
    int base = 3;
    for (int l = 0; l < 3; ++l) {
        w1[l] = (const float*)d_in[base + 0];
        b1[l] = (const float*)d_in[base + 1];
        w2[l] = (const float*)d_in[base + 2];
        b2[l] = (const float*)d_in[base + 3];
        gm[l] = (const float*)d_in[base + 4];
        bt[l] = (const float*)d_in[base + 5];
        base += 6;
    }
    const float* fcw = (const float*)d_in[21];
    const float* fcb = (const float*)d_in[22];

    const int F = 128;
    const int N = in_sizes[0] / F;
    const int E = in_sizes[1] / 2;
    const int G = out_size;

    // workspace layout (floats): agg[N*128] | h[N*64] | stats[128] | pooled[G*64]
    float* agg    = (float*)d_ws;
    float* h      = agg + (size_t)N * F;
    float* stats  = h + (size_t)N * HID;
    float* pooled = stats + 128;

    const int tiles = (N + 15) / 16;
    const long long nh = (long long)N * HID;
    const unsigned nhBlocks = (unsigned)((nh + 255) / 256);

    // ---------------- layer 1 (FIN = 128) ----------------
    hipMemcpyAsync(agg, x, (size_t)N * F * sizeof(float), hipMemcpyDeviceToDevice, stream);
    {
        long long tot = (long long)E * (F / 4);
        edge_scatter<<<(unsigned)((tot + 255) / 256), 256, 0, stream>>>(x, agg, ei, E, F, 5);
    }
    hipMemsetAsync(stats, 0, 128 * sizeof(float), stream);
    mlp_wmma<128><<<tiles, 32, 0, stream>>>(agg, w1[0], b1[0], w2[0], b2[0], h, N);
    bn_stats<<<256, 256, 0, stream>>>(h, stats, N);
    bn_apply<<<nhBlocks, 256, 0, stream>>>(h, stats, gm[0], bt[0], N);

    // ---------------- layers 2,3 (FIN = 64) ----------------
    for (int l = 1; l < 3; ++l) {
        hipMemcpyAsync(agg, h, (size_t)N * HID * sizeof(float), hipMemcpyDeviceToDevice, stream);
        long long tot = (long long)E * (HID / 4);
        edge_scatter<<<(unsigned)((tot + 255) / 256), 256, 0, stream>>>(h, agg, ei, E, HID, 4);
        hipMemsetAsync(stats, 0, 128 * sizeof(float), stream);
        mlp_wmma<64><<<tiles, 32, 0, stream>>>(agg, w1[l], b1[l], w2[l], b2[l], h, N);
        bn_stats<<<256, 256, 0, stream>>>(h, stats, N);
        bn_apply<<<nhBlocks, 256, 0, stream>>>(h, stats, gm[l], bt[l], N);
    }

    // ---------------- readout ----------------
    hipMemsetAsync(pooled, 0, (size_t)G * HID * sizeof(float), stream);
    pool_scatter<<<nhBlocks, 256, 0, stream>>>(h, batch, pooled, N);
    pool_gemv<<<(G + 255) / 256, 256, 0, stream>>>(pooled, fcw, fcb, (float*)d_out, G);
}